// QLSTM_65481071402890
// MI455X (gfx1250) — compile-verified
//
#include <hip/hip_runtime.h>
#include <math.h>

typedef __attribute__((ext_vector_type(2))) float v2f;
typedef __attribute__((ext_vector_type(4))) float v4f;
typedef __attribute__((ext_vector_type(8))) float v8f;

#define TVAL 256
#define BVAL 2048
#define DVAL 128
#define DH   132   // D + H

// ---------------------------------------------------------------------------
// Phase 1: xproj[(t*B+b)*16 + n] = sum_{k<128} x[t,b,k] * W_{gate(n)}[j(n)][k]
// GEMM (T*B x 128) x (128 x 16) via v_wmma_f32_16x16x4_f32.
//
// K-remapping (sum over K is order-invariant, so we choose our own slot->k
// map, identical for A and B):
//   lane (m = lane&15, hi = lane>>4) holds float4 chunk q = x[m][64*hi+4q..+3]
//   frag (q, even) = chunk.xy ; frag (q, odd) = chunk.zw
// B panel loaded with the same map (n = lane&15 selects the weight row).
// D layout: vgpr r, lane L -> D[r + 8*(L>>4)][L&15]
//
// sched_barrier(0) forces all 16 b128 loads of a tile to issue before the
// WMMA chain starts -> 16 outstanding loads per wave (vs 2 when the
// scheduler interleaves), which is what HBM at 23.3 TB/s needs.
// ---------------------------------------------------------------------------
__global__ void __launch_bounds__(256)
qlstm_xproj_wmma(const float* __restrict__ x,
                 const float* __restrict__ Wf, const float* __restrict__ Wi,
                 const float* __restrict__ Wu, const float* __restrict__ Wo,
                 float* __restrict__ xproj, int numTiles)
{
    const int lane   = threadIdx.x & 31;
    const int wid    = (blockIdx.x * blockDim.x + threadIdx.x) >> 5;
    const int nWaves = (gridDim.x * blockDim.x) >> 5;
    const int n      = lane & 15;
    const int hi     = lane >> 4;

    // per-lane weight row pointer (divergent select, stays in VGPRs)
    const int gate = n >> 2;
    const float* Wr = (gate == 0) ? Wf : (gate == 1) ? Wi : (gate == 2) ? Wu : Wo;
    Wr += (n & 3) * DH + 64 * hi;

    // Load the whole 16x128 weight panel as 16 float4 chunks (64 VGPRs).
    v4f wf[16];
#pragma unroll
    for (int q = 0; q < 16; ++q)
        wf[q] = *(const v4f*)(Wr + 4 * q);

    for (int tile = wid; tile < numTiles; tile += nWaves) {
        // Prefetch next tile's half-row for this lane (2 x 128B lines).
        if (tile + nWaves < numTiles) {
            const char* nx = (const char*)(x
                + (size_t)((tile + nWaves) * 16 + n) * DVAL + 64 * hi);
            __builtin_prefetch(nx, 0, 3);
            __builtin_prefetch(nx + 128, 0, 3);
        }
        __builtin_amdgcn_sched_barrier(0);   // prefetches first

        const float* xr = x + (size_t)(tile * 16 + n) * DVAL + 64 * hi;
        v4f a4[16];
#pragma unroll
        for (int q = 0; q < 16; ++q)
            a4[q] = *(const v4f*)(xr + 4 * q);

        __builtin_amdgcn_sched_barrier(0);   // all 16 loads in flight first

        v8f acc0 = {0.f, 0.f, 0.f, 0.f, 0.f, 0.f, 0.f, 0.f};
        v8f acc1 = {0.f, 0.f, 0.f, 0.f, 0.f, 0.f, 0.f, 0.f};
#pragma unroll
        for (int q = 0; q < 16; ++q) {
            v2f ae = {a4[q][0], a4[q][1]};
            v2f ao = {a4[q][2], a4[q][3]};
            v2f be = {wf[q][0], wf[q][1]};
            v2f bo = {wf[q][2], wf[q][3]};
            acc0 = __builtin_amdgcn_wmma_f32_16x16x4_f32(
                false, ae, false, be, (short)0, acc0, false, false);
            acc1 = __builtin_amdgcn_wmma_f32_16x16x4_f32(
                false, ao, false, bo, (short)0, acc1, false, false);
        }
        v8f acc = acc0 + acc1;

        float* op = xproj + (size_t)tile * 256 + n;
#pragma unroll
        for (int r = 0; r < 8; ++r)
            op[(r + 8 * hi) * 16] = acc[r];
    }
}

// ---------------------------------------------------------------------------
// Phase 2: sequential recurrence, one thread per batch element.
// arg[n] = xproj[n] + (b+th)[n] + sum_k h[k]*Wh[n][k];  C[n] = cos(arg[n])
// gate outputs: out0=C1C2C3, out1=C0C1, out2=C0C1C2, out3=C0C1C2C3
// ---------------------------------------------------------------------------
__device__ __forceinline__ float sigm(float v) { return 1.f / (1.f + expf(-v)); }

__global__ void __launch_bounds__(64)
qlstm_recur(const float* __restrict__ xproj,
            const float* __restrict__ Wf, const float* __restrict__ bf,
            const float* __restrict__ Wi, const float* __restrict__ bi,
            const float* __restrict__ Wu, const float* __restrict__ bu,
            const float* __restrict__ Wo, const float* __restrict__ bo,
            const float* __restrict__ thf, const float* __restrict__ thi,
            const float* __restrict__ thu, const float* __restrict__ tho,
            float* __restrict__ out)
{
    const int b = blockIdx.x * blockDim.x + threadIdx.x;
    if (b >= BVAL) return;

    float wh[16][4];
    float bth[16];
#pragma unroll
    for (int g = 0; g < 4; ++g) {
        const float* W  = (g == 0) ? Wf : (g == 1) ? Wi : (g == 2) ? Wu : Wo;
        const float* bb = (g == 0) ? bf : (g == 1) ? bi : (g == 2) ? bu : bo;
        const float* th = (g == 0) ? thf : (g == 1) ? thi : (g == 2) ? thu : tho;
#pragma unroll
        for (int j = 0; j < 4; ++j) {
            bth[g * 4 + j] = bb[j] + th[j];
#pragma unroll
            for (int k = 0; k < 4; ++k)
                wh[g * 4 + j][k] = W[j * DH + 128 + k];
        }
    }

    float h[4] = {0.f, 0.f, 0.f, 0.f};
    float c[4] = {0.f, 0.f, 0.f, 0.f};

    const float4* xp4 = (const float4*)xproj;
    float4 cur[4], nxt[4];
#pragma unroll
    for (int q = 0; q < 4; ++q) cur[q] = xp4[(size_t)b * 4 + q];

#pragma unroll 1
    for (int t = 0; t < TVAL; ++t) {
        if (t + 1 < TVAL) {
            const float4* p = xp4 + (size_t)((t + 1) * BVAL + b) * 4;
#pragma unroll
            for (int q = 0; q < 4; ++q) nxt[q] = p[q];
        }
        float xp[16] = {cur[0].x, cur[0].y, cur[0].z, cur[0].w,
                        cur[1].x, cur[1].y, cur[1].z, cur[1].w,
                        cur[2].x, cur[2].y, cur[2].z, cur[2].w,
                        cur[3].x, cur[3].y, cur[3].z, cur[3].w};
        float C[16];
#pragma unroll
        for (int n = 0; n < 16; ++n) {
            float arg = xp[n] + bth[n]
                      + h[0] * wh[n][0] + h[1] * wh[n][1]
                      + h[2] * wh[n][2] + h[3] * wh[n][3];
            C[n] = cosf(arg);
        }
        float gv[4][4];
#pragma unroll
        for (int g = 0; g < 4; ++g) {
            const float c0 = C[g * 4 + 0], c1 = C[g * 4 + 1];
            const float c2 = C[g * 4 + 2], c3 = C[g * 4 + 3];
            gv[g][0] = c1 * c2 * c3;
            gv[g][1] = c0 * c1;
            gv[g][2] = c0 * c1 * c2;
            gv[g][3] = gv[g][2] * c3;
        }
#pragma unroll
        for (int w = 0; w < 4; ++w) {
            const float fw = sigm(gv[0][w]);
            const float iw = sigm(gv[1][w]);
            const float uw = tanhf(gv[2][w]);
            const float ow = sigm(gv[3][w]);
            c[w] = fw * c[w] + iw * uw;
            h[w] = ow * tanhf(c[w]);
        }
        ((float4*)out)[(size_t)t * BVAL + b] = make_float4(h[0], h[1], h[2], h[3]);
#pragma unroll
        for (int q = 0; q < 4; ++q) cur[q] = nxt[q];
    }

    float4* tail = (float4*)(out + (size_t)TVAL * BVAL * 4);
    tail[b]        = make_float4(h[0], h[1], h[2], h[3]);
    tail[BVAL + b] = make_float4(c[0], c[1], c[2], c[3]);
}

// ---------------------------------------------------------------------------
extern "C" void kernel_launch(void* const* d_in, const int* in_sizes, int n_in,
                              void* d_out, int out_size, void* d_ws, size_t ws_size,
                              hipStream_t stream)
{
    const float* x   = (const float*)d_in[0];
    const float* Wf  = (const float*)d_in[1];
    const float* bf  = (const float*)d_in[2];
    const float* Wi  = (const float*)d_in[3];
    const float* bi  = (const float*)d_in[4];
    const float* Wu  = (const float*)d_in[5];
    const float* bu  = (const float*)d_in[6];
    const float* Wo  = (const float*)d_in[7];
    const float* bo  = (const float*)d_in[8];
    const float* thf = (const float*)d_in[9];
    const float* thi = (const float*)d_in[10];
    const float* thu = (const float*)d_in[11];
    const float* tho = (const float*)d_in[12];
    float* out   = (float*)d_out;
    float* xproj = (float*)d_ws;   // needs T*B*16*4 = 32 MB

    const int numTiles = (TVAL * BVAL) / 16;  // 32768 16-row tiles
    qlstm_xproj_wmma<<<1024, 256, 0, stream>>>(x, Wf, Wi, Wu, Wo, xproj, numTiles);
    qlstm_recur<<<BVAL / 64, 64, 0, stream>>>(xproj, Wf, bf, Wi, bi, Wu, bu,
                                              Wo, bo, thf, thi, thu, tho, out);
}